// SAGE_14748917695089
// MI455X (gfx1250) — compile-verified
//
#include <hip/hip_runtime.h>

#define N_NODES 50000
#define N_EDGES 800000

typedef __attribute__((ext_vector_type(2))) float v2f;
typedef __attribute__((ext_vector_type(8))) float v8f;

// ---------------------------------------------------------------- utilities
__global__ void zero_f32(float* __restrict__ p, int n) {
  int i = blockIdx.x * blockDim.x + threadIdx.x;
  if (i < n) p[i] = 0.0f;
}

__global__ void degree_kernel(const int* __restrict__ dst, float* __restrict__ deg, int E) {
  int e = blockIdx.x * blockDim.x + threadIdx.x;
  if (e < E) atomicAdd(&deg[dst[e]], 1.0f);
}

// One wave32 per edge; each lane owns a float2 of the feature row
// (coalesced global_load_b64), scalar f32 atomics into the destination row.
__global__ void scatter_kernel(const float* __restrict__ h, const int* __restrict__ src,
                               const int* __restrict__ dst, float* __restrict__ agg,
                               int E, int d) {
  int gid = blockIdx.x * blockDim.x + threadIdx.x;
  int e = gid >> 5;
  int lane = gid & 31;
  if (e >= E) return;
  int s = src[e];
  int t = dst[e];
  const v2f* hs = (const v2f*)(h + (size_t)s * d);
  float* at = agg + (size_t)t * d;
  int half_d = d >> 1;
  for (int f = lane; f < half_d; f += 32) {
    v2f v = hs[f];
    atomicAdd(&at[2 * f + 0], v[0]);
    atomicAdd(&at[2 * f + 1], v[1]);
  }
}

// h_neigh = agg / max(deg, 1)  (in place)
__global__ void scale_kernel(float* __restrict__ agg, const float* __restrict__ deg,
                             int n, int d) {
  int i = blockIdx.x * blockDim.x + threadIdx.x;
  if (i >= n * d) return;
  int node = i / d;
  float dg = deg[node];
  agg[i] *= 1.0f / fmaxf(dg, 1.0f);
}

// Pack W[K x N] into pair-major layout: Wp[p*N + col] = {W[2p][col], W[2p+1][col]}
// so the WMMA B-operand (lane needs K=k,k+1 at one column) is a single b64 load.
__global__ void pack_weights(const float* __restrict__ W, float* __restrict__ Wp,
                             int K, int N) {
  int i = blockIdx.x * blockDim.x + threadIdx.x;  // over (K/2)*N
  int total = (K >> 1) * N;
  if (i >= total) return;
  int p = i / N;
  int col = i - p * N;
  Wp[2 * i + 0] = W[(size_t)(2 * p + 0) * N + col];
  Wp[2 * i + 1] = W[(size_t)(2 * p + 1) * N + col];
}

// ------------------------------------------------------------- WMMA GEMM
// out[M x N] = relu?( A0[MxK] @ W0 + A1[MxK] @ W1 + bias[N] ), W* pre-packed.
// One wave per 16-row strip; the wave keeps N/16 accumulator tiles so each
// A-operand load (float2/lane) feeds N/16 WMMAs. Compile-time K,N give
// immediate-offset addressing. fp32 V_WMMA_F32_16X16X4_F32 throughout.
// A 16x4 f32 layout: lanes 0-15 carry K={k,k+1}, lanes 16-31 K={k+2,k+3},
// M = lane&15. C/D: lane<16 -> (M=r,N=lane), lane>=16 -> (M=r+8,N=lane-16).
template <int K, int N>
__global__ void sage_gemm_wmma(const float* __restrict__ A0, const float* __restrict__ A1,
                               const float* __restrict__ W0p, const float* __restrict__ W1p,
                               const float* __restrict__ bias, float* __restrict__ out,
                               int do_relu) {
  constexpr int NT = N / 16;
  int wave = (blockIdx.x * blockDim.x + threadIdx.x) >> 5;
  constexpr int MTILES = N_NODES / 16;  // 3125, exact
  if (wave >= MTILES) return;           // wave-uniform: EXEC all-1s for WMMA

  int m0 = wave << 4;
  unsigned lane = threadIdx.x & 31u;
  unsigned half = lane >> 4;  // 0 = lanes 0-15, 1 = lanes 16-31
  unsigned l16  = lane & 15u;

  v8f acc[NT];
#pragma unroll
  for (int t = 0; t < NT; ++t) {
    float bval = bias[t * 16 + l16];
#pragma unroll
    for (int r = 0; r < 8; ++r) acc[t][r] = bval;
  }

  const float* a0row = A0 + (size_t)(m0 + l16) * K;
  const float* a1row = A1 + (size_t)(m0 + l16) * K;

  for (int k0 = 0; k0 < K; k0 += 4) {
    int kk = k0 + 2 * (int)half;
    int p  = (k0 >> 1) + (int)half;  // float2-pair row index
    v2f a = *(const v2f*)(a0row + kk);  // 8B aligned (kk even)
#pragma unroll
    for (int t = 0; t < NT; ++t) {
      v2f b = *(const v2f*)(W0p + 2 * ((size_t)p * N + t * 16 + l16));
      acc[t] = __builtin_amdgcn_wmma_f32_16x16x4_f32(false, a, false, b, (short)0,
                                                     acc[t], false, false);
    }
  }
  for (int k0 = 0; k0 < K; k0 += 4) {
    int kk = k0 + 2 * (int)half;
    int p  = (k0 >> 1) + (int)half;
    v2f a = *(const v2f*)(a1row + kk);
#pragma unroll
    for (int t = 0; t < NT; ++t) {
      v2f b = *(const v2f*)(W1p + 2 * ((size_t)p * N + t * 16 + l16));
      acc[t] = __builtin_amdgcn_wmma_f32_16x16x4_f32(false, a, false, b, (short)0,
                                                     acc[t], false, false);
    }
  }

  if (do_relu) {
#pragma unroll
    for (int t = 0; t < NT; ++t)
#pragma unroll
      for (int r = 0; r < 8; ++r) acc[t][r] = fmaxf(acc[t][r], 0.0f);
  }

  int rbase = m0 + (half ? 8 : 0);
#pragma unroll
  for (int t = 0; t < NT; ++t)
#pragma unroll
    for (int r = 0; r < 8; ++r)
      out[(size_t)(rbase + r) * N + t * 16 + l16] = acc[t][r];
}

// ------------------------------------------------------------- launcher
extern "C" void kernel_launch(void* const* d_in, const int* in_sizes, int n_in,
                              void* d_out, int out_size, void* d_ws, size_t ws_size,
                              hipStream_t stream) {
  (void)in_sizes; (void)n_in; (void)out_size; (void)ws_size;

  const float* x   = (const float*)d_in[0];
  const int*   src = (const int*)d_in[1];
  const int*   dst = (const int*)d_in[2];
  const float* Ws0 = (const float*)d_in[3];
  const float* Wn0 = (const float*)d_in[4];
  const float* b0  = (const float*)d_in[5];
  const float* Ws1 = (const float*)d_in[6];
  const float* Wn1 = (const float*)d_in[7];
  const float* b1  = (const float*)d_in[8];
  const float* Ws2 = (const float*)d_in[9];
  const float* Wn2 = (const float*)d_in[10];
  const float* b2  = (const float*)d_in[11];
  float* out = (float*)d_out;

  // workspace: deg | agg(N*128) | h1(N*128) | h2(N*128) | w0p(128*128) | w1p(128*128)
  float* deg = (float*)d_ws;
  float* agg = deg + ((N_NODES + 63) & ~63);
  float* h1  = agg + (size_t)N_NODES * 128;
  float* h2  = h1 + (size_t)N_NODES * 128;
  float* w0p = h2 + (size_t)N_NODES * 128;
  float* w1p = w0p + 128 * 128;

  const int B = 256;
  const int wpb = B / 32;

  // degrees (shared by all layers)
  zero_f32<<<(N_NODES + B - 1) / B, B, 0, stream>>>(deg, N_NODES);
  degree_kernel<<<(N_EDGES + B - 1) / B, B, 0, stream>>>(dst, deg, N_EDGES);

  constexpr int MT = N_NODES / 16;
  const int gemm_blocks = (MT + wpb - 1) / wpb;
  const long sthreads = (long)N_EDGES * 32;
  const int sblocks = (int)((sthreads + B - 1) / B);

#define SAGE_LAYER(hin, Kc, Nc, Wself, Wneigh, bias, hout, relu)                       \
  do {                                                                                 \
    int nd = N_NODES * (Kc);                                                           \
    int pk = ((Kc) / 2) * (Nc);                                                        \
    zero_f32<<<(nd + B - 1) / B, B, 0, stream>>>(agg, nd);                             \
    pack_weights<<<(pk + B - 1) / B, B, 0, stream>>>(Wself, w0p, (Kc), (Nc));          \
    pack_weights<<<(pk + B - 1) / B, B, 0, stream>>>(Wneigh, w1p, (Kc), (Nc));         \
    scatter_kernel<<<sblocks, B, 0, stream>>>(hin, src, dst, agg, N_EDGES, (Kc));      \
    scale_kernel<<<(nd + B - 1) / B, B, 0, stream>>>(agg, deg, N_NODES, (Kc));         \
    sage_gemm_wmma<Kc, Nc><<<gemm_blocks, B, 0, stream>>>(hin, agg, w0p, w1p, bias,    \
                                                          hout, relu);                 \
  } while (0)

  SAGE_LAYER(x,  64, 128, Ws0, Wn0, b0, h1, 1);
  SAGE_LAYER(h1, 128, 128, Ws1, Wn1, b1, h2, 1);
  SAGE_LAYER(h2, 128, 64, Ws2, Wn2, b2, out, 0);
#undef SAGE_LAYER
}